// SoftTruncatedGaussianMixtureAnalysis_90391881712222
// MI455X (gfx1250) — compile-verified
//
#include <hip/hip_runtime.h>
#include <math.h>

// C=5 classes, K=4 components, D=32 dims, N=100000 samples (N taken from in_sizes)
#define C_CLS 5
#define K_CMP 4
#define D_DIM 32
#define CKPAD 32   // 20 real (c,k) columns padded to 32 for two 16-wide WMMA tiles

// accurate softplus (used only in the tiny precompute kernel)
static __device__ __forceinline__ float softplusf(float x) {
    return fmaxf(x, 0.0f) + log1pf(expf(-fabsf(x)));
}
// fast softplus on v_exp_f32 / v_log_f32 (hot loop: ~26M evaluations)
static __device__ __forceinline__ float softplus_fast(float x) {
    return fmaxf(x, 0.0f) + __logf(1.0f + __expf(-fabsf(x)));
}
static __device__ __forceinline__ float ndtrf(float x) {
    return 0.5f * erfcf(-x * 0.70710678118654752440f);
}

typedef __attribute__((ext_vector_type(2))) float v2f;
typedef __attribute__((ext_vector_type(8))) float v8f;

__constant__ float kEPS    = 1.1920928955078125e-07f;
__constant__ float kLOG2PI = 1.8378770664093453f;

// ---------------- kernel 0: zero workspace ----------------
__global__ void stgma_zero(float* ws, int n) {
    int i = blockIdx.x * blockDim.x + threadIdx.x;
    if (i < n) ws[i] = 0.0f;
}

// ---------------- kernel 1: per-(c,k) precompute ----------------
// grid = C*K blocks, 32 threads (one wave handles the D=32 dims of one (c,k))
__global__ void stgma_precompute(const float* __restrict__ mu,
                                 const float* __restrict__ lower,
                                 const float* __restrict__ upper,
                                 const float* __restrict__ sigma,
                                 const float* __restrict__ logits,
                                 float* __restrict__ W,     // [64][CKPAD]
                                 float* __restrict__ cst) { // [CKPAD]
    int ck = blockIdx.x;             // 0..19
    int c  = ck / K_CMP;
    int d  = threadIdx.x;            // 0..31
    int idx = ck * D_DIM + d;

    float m     = mu[idx];
    float scale = kEPS + softplusf(sigma[idx]);
    float invs  = 1.0f / scale;
    float inv2  = invs * invs;

    float z = ndtrf((upper[idx] - m) * invs) - ndtrf((lower[idx] - m) * invs);
    // per-dim constant: -0.5*m^2/s^2 - log(s)
    float cd = -0.5f * m * m * inv2 - logf(scale);

    float zp = z, cs = cd;
    #pragma unroll
    for (int off = 16; off > 0; off >>= 1) {
        zp *= __shfl_xor(zp, off, 32);
        cs += __shfl_xor(cs, off, 32);
    }

    // weight matrix rows: [0..31] linear coef (mu/s^2), [32..63] quadratic (-0.5/s^2)
    W[d * CKPAD + ck]            = m * inv2;
    W[(D_DIM + d) * CKPAD + ck]  = -0.5f * inv2;

    if (d == 0) {
        float mx = -INFINITY;
        for (int kk = 0; kk < K_CMP; ++kk) mx = fmaxf(mx, logits[c * K_CMP + kk]);
        float se = 0.0f;
        for (int kk = 0; kk < K_CMP; ++kk) se += expf(logits[c * K_CMP + kk] - mx);
        float cat_lp = logits[ck] - (mx + logf(se));
        cst[ck] = cat_lp - logf(zp + kEPS) + cs - 0.5f * kLOG2PI * (float)D_DIM;
    }
}

// ---------------- kernel 2: main (async LDS staging + WMMA GEMM + masked finalize) ----------------
// 256 threads (8 wave32), 128 rows of X per block.
__global__ __launch_bounds__(256)
void stgma_main(const float* __restrict__ X,      // [N][32]
                const int*   __restrict__ y,      // [N]
                const float* __restrict__ resp,   // [C][N][K]
                const float* __restrict__ lower,  // [C][K][32]
                const float* __restrict__ upper,  // [C][K][32]
                const float* __restrict__ etaP,
                const float* __restrict__ W,      // [64][CKPAD]
                const float* __restrict__ cst,    // [CKPAD]
                float* __restrict__ Sg,           // [C] atomic accum
                float* __restrict__ Cg,           // [C] atomic counts
                int N) {
    __shared__ float Xs[128][D_DIM];    // 16 KB
    __shared__ float Ws[64][CKPAD];     // 8 KB
    __shared__ float Gs[128][CKPAD];    // 16 KB
    __shared__ float blkS[C_CLS];
    __shared__ float blkC[C_CLS];

    const int tid = threadIdx.x;
    const int n0  = blockIdx.x * 128;

    // ---- stage X tile straight into LDS via async copies (ASYNCcnt path) ----
    // 1024 16-byte chunks: 4 per thread.
    for (int i = tid; i < 1024; i += 256) {
        int row = i >> 3;
        int q   = i & 7;
        int n   = n0 + row;
        unsigned lds_off = (unsigned)(unsigned long long)(&Xs[row][q * 4]);
        if (n < N) {
            const float* gp = X + (size_t)n * D_DIM + q * 4;
            asm volatile("global_load_async_to_lds_b128 %0, %1, off"
                         :: "v"(lds_off), "v"(gp)
                         : "memory");
        } else {
            float4 zz = make_float4(0.f, 0.f, 0.f, 0.f);
            *(float4*)&Xs[row][q * 4] = zz;   // zero-pad tail rows
        }
    }

    // stage weights into LDS (small, synchronous)
    for (int i = tid; i < 64 * CKPAD; i += 256)
        (&Ws[0][0])[i] = W[i];

    if (tid < C_CLS) { blkS[tid] = 0.0f; blkC[tid] = 0.0f; }

    asm volatile("s_wait_asynccnt 0" ::: "memory");
    __syncthreads();

    // ---- WMMA phase: G[n, ck] = Σ_j feat[n,j] * W[j,ck], j over augmented K=64 ----
    const int wave = tid >> 5;
    const int lane = tid & 31;
    const int half = lane >> 4;       // 0: lanes 0-15, 1: lanes 16-31
    const int mrow = lane & 15;
    const int r0   = wave * 16;       // this wave's 16-row tile

    #pragma unroll
    for (int t = 0; t < 2; ++t) {     // two 16-wide column tiles (ck 0-15, 16-31)
        v8f acc = {0.f, 0.f, 0.f, 0.f, 0.f, 0.f, 0.f, 0.f};
        const int col = t * 16 + mrow;
        #pragma unroll
        for (int kk = 0; kk < 64; kk += 4) {
            const int j0 = kk + half * 2;   // A lanes 0-15 hold K=kk,kk+1; 16-31 hold kk+2,kk+3
            float a0, a1;
            if (kk < D_DIM) {               // linear features: x
                a0 = Xs[r0 + mrow][j0];
                a1 = Xs[r0 + mrow][j0 + 1];
            } else {                        // quadratic features: x^2
                float x0 = Xs[r0 + mrow][j0 - D_DIM];
                float x1 = Xs[r0 + mrow][j0 + 1 - D_DIM];
                a0 = x0 * x0;
                a1 = x1 * x1;
            }
            v2f A = {a0, a1};
            v2f B = {Ws[j0][col], Ws[j0 + 1][col]};   // B rows follow same K striping
            acc = __builtin_amdgcn_wmma_f32_16x16x4_f32(
                /*neg_a=*/false, A, /*neg_b=*/false, B,
                /*c_mod=*/(short)0, acc, /*reuse_a=*/false, /*reuse_b=*/false);
        }
        #pragma unroll
        for (int v = 0; v < 8; ++v)                    // C/D: vgpr v -> M = v + 8*half
            Gs[r0 + v + half * 8][col] = acc[v];
    }
    __syncthreads();

    // ---- finalize: only class c = y[n] contributes; add soft-truncation terms ----
    const float eta = *etaP;
    for (int it = tid; it < 128 * K_CMP; it += 256) {
        int row = it >> 2;
        int k   = it & 3;
        int n   = n0 + row;
        if (n < N) {
            int c  = y[n];
            int ck = c * K_CMP + k;
            const float* lo = lower + (size_t)ck * D_DIM;
            const float* up = upper + (size_t)ck * D_DIM;
            float lu = 0.0f;
            #pragma unroll 4
            for (int d = 0; d < D_DIM; ++d) {
                float x = Xs[row][d];
                lu -= softplus_fast(-eta * (x - lo[d]));
                lu -= softplus_fast(-eta * (up[d] - x));
            }
            float logpdf = Gs[row][ck] + cst[ck] + lu;
            float val = resp[((size_t)c * N + n) * K_CMP + k] * logpdf;
            atomicAdd(&blkS[c], val);
            if (k == 0) atomicAdd(&blkC[c], 1.0f);
        }
    }
    __syncthreads();
    if (tid < C_CLS) {
        atomicAdd(&Sg[tid], blkS[tid]);
        atomicAdd(&Cg[tid], blkC[tid]);
    }
}

// ---------------- kernel 3: final scalar ----------------
__global__ void stgma_finalize(const float* __restrict__ Sg,
                               const float* __restrict__ Cg,
                               float* __restrict__ out) {
    if (threadIdx.x == 0 && blockIdx.x == 0) {
        float s = 0.0f;
        for (int c = 0; c < C_CLS; ++c)
            s += Sg[c] / (Cg[c] * (float)K_CMP);
        out[0] = -s;
    }
}

extern "C" void kernel_launch(void* const* d_in, const int* in_sizes, int n_in,
                              void* d_out, int out_size, void* d_ws, size_t ws_size,
                              hipStream_t stream) {
    const float* X      = (const float*)d_in[0];
    const int*   y      = (const int*)  d_in[1];
    const float* resp   = (const float*)d_in[2];
    const float* mu     = (const float*)d_in[3];
    const float* lower  = (const float*)d_in[4];
    const float* upper  = (const float*)d_in[5];
    const float* sigma  = (const float*)d_in[6];
    const float* logits = (const float*)d_in[7];
    const float* eta    = (const float*)d_in[8];
    const int N = in_sizes[1];

    float* ws  = (float*)d_ws;
    float* W   = ws;          // 64*32 = 2048 floats
    float* cst = ws + 2048;   // 32 floats
    float* Sg  = ws + 2080;   // 8 floats
    float* Cg  = ws + 2088;   // 8 floats
    const int zn = 2096;

    stgma_zero<<<(zn + 255) / 256, 256, 0, stream>>>(ws, zn);
    stgma_precompute<<<C_CLS * K_CMP, D_DIM, 0, stream>>>(mu, lower, upper, sigma, logits, W, cst);
    const int nb = (N + 127) / 128;
    stgma_main<<<nb, 256, 0, stream>>>(X, y, resp, lower, upper, eta, W, cst, Sg, Cg, N);
    stgma_finalize<<<1, 32, 0, stream>>>(Sg, Cg, (float*)d_out);
}